// TSTabFusionMLM_66975720013983
// MI455X (gfx1250) — compile-verified
//
#include <hip/hip_runtime.h>
#include <stdint.h>

// ---- reference constants ----
#define MASK_P    0.15f
#define REPLACE_P 0.80f
#define RANDOM_P  0.10f

typedef float v2f __attribute__((ext_vector_type(2)));
typedef float v8f __attribute__((ext_vector_type(8)));

// =====================================================================
// Kernel 1: time-series masking (shared by x_ts [C=128] and x_ts_cat [C=256]).
// Grid: (T/1024, C/32, BS), 256 threads.  Each block:
//   - async-stages u_mask/u_act/r_idx for its 1024-wide t-tile into LDS
//   - decodes to packed code word: (swap_idx << 2) | {0=keep,1=zero,2=swap}
//   - streams 32 channel rows as float4 (b128 load/store), applying the code
// =====================================================================
__global__ __launch_bounds__(256)
void mask_ts_kernel(const float* __restrict__ x,
                    const float* __restrict__ u_mask,
                    const float* __restrict__ u_act,
                    const int*   __restrict__ r_idx,
                    float* __restrict__ out,
                    int C, int T)
{
    constexpr int TPOS = 1024;   // t positions per block
    constexpr int CG   = 32;     // channels per block

    __shared__ float s_mask[TPOS];
    __shared__ float s_act [TPOS];
    __shared__ int   s_idx [TPOS];
    __shared__ int   s_code[TPOS];

    const int t0  = blockIdx.x * TPOS;
    const int c0  = blockIdx.y * CG;
    const int b   = blockIdx.z;
    const int tid = threadIdx.x;

    // ---- stage metadata with gfx1250 async global->LDS loads ----
    {
        const float* gm = u_mask + (size_t)b * T + t0;
        const float* ga = u_act  + (size_t)b * T + t0;
        const int*   gi = r_idx  + (size_t)b * T + t0;
#pragma unroll
        for (int k = 0; k < TPOS / 256; ++k) {
            const int i = tid + k * 256;
            unsigned dm = (unsigned)(uintptr_t)(&s_mask[i]);
            unsigned da = (unsigned)(uintptr_t)(&s_act [i]);
            unsigned di = (unsigned)(uintptr_t)(&s_idx [i]);
            asm volatile("global_load_async_to_lds_b32 %0, %1, off"
                         :: "v"(dm), "v"(gm + i) : "memory");
            asm volatile("global_load_async_to_lds_b32 %0, %1, off"
                         :: "v"(da), "v"(ga + i) : "memory");
            asm volatile("global_load_async_to_lds_b32 %0, %1, off"
                         :: "v"(di), "v"(gi + i) : "memory");
        }
        asm volatile("s_wait_asynccnt 0" ::: "memory");
    }
    __syncthreads();

    // ---- decode decision per t position ----
#pragma unroll
    for (int k = 0; k < TPOS / 256; ++k) {
        const int i = tid + k * 256;
        const float um = s_mask[i];
        const float ua = s_act [i];
        int code = 0;
        if (um < MASK_P) {
            if (ua < REPLACE_P)                 code = 1;  // replace -> zero
            else if (ua < REPLACE_P + RANDOM_P) code = 2;  // swap
        }
        s_code[i] = (s_idx[i] << 2) | code;
    }
    __syncthreads();

    // ---- stream channel rows: each thread owns one float4 of the t-tile ----
    const int tt = tid * 4;
#define APPLY_MASK(comp, e)                                   \
    { const int w = s_code[tt + (e)]; const int cd = w & 3;   \
      if (cd == 1) (comp) = 0.0f;                             \
      else if (cd == 2) (comp) = xrow[w >> 2]; }

    for (int c = 0; c < CG; ++c) {
        const float* __restrict__ xrow = x   + ((size_t)b * C + (c0 + c)) * T;
        float*       __restrict__ orow = out + ((size_t)b * C + (c0 + c)) * T;
        float4 v = *(const float4*)(xrow + t0 + tt);
        APPLY_MASK(v.x, 0)
        APPLY_MASK(v.y, 1)
        APPLY_MASK(v.z, 2)
        APPLY_MASK(v.w, 3)
        *(float4*)(orow + t0 + tt) = v;
    }
#undef APPLY_MASK
}

// =====================================================================
// Kernel 2: static categorical masking (64x32 int32 -> float out)
// =====================================================================
__global__ void mask_cat_kernel(const int* __restrict__ x_cat,
                                const float* __restrict__ um,
                                const float* __restrict__ ua,
                                const int* __restrict__ rv,
                                float* __restrict__ out, int n)
{
    const int i = blockIdx.x * blockDim.x + threadIdx.x;
    if (i >= n) return;
    int v = x_cat[i];
    const float m = um[i], a = ua[i];
    if (m < MASK_P) {
        if (a < REPLACE_P)                 v = 0;
        else if (a < REPLACE_P + RANDOM_P) v = rv[i];
    }
    out[i] = (float)v;
}

// =====================================================================
// Kernel 3: static continuous masking (64x64).
// Column means computed on the matrix core:
//   means = (1/64) * ones(16x4) x  X_chunk(4x16), accumulated over 16 chunks.
// One wave per 16-column group (4 waves, 128 threads, single block).
// Since A is a constant matrix, the K-row ordering inside B is irrelevant.
// =====================================================================
__global__ __launch_bounds__(128)
void mask_cont_kernel(const float* __restrict__ x,
                      const float* __restrict__ um,
                      const float* __restrict__ ua,
                      const int*   __restrict__ ridx,
                      float* __restrict__ out)
{
    constexpr int BS = 64, NC = 64;
    __shared__ float s_mean[NC];

    const int tid     = threadIdx.x;
    const int lane    = tid & 31;
    const int wave    = tid >> 5;          // 0..3
    const int colbase = wave * 16;
    const int ncol    = lane & 15;         // N position within tile
    const int half    = lane >> 4;         // lane half selects K rows

    v2f a; a.x = 1.0f / 64.0f; a.y = 1.0f / 64.0f;   // constant A (all 1/64)
    v8f c = {};
    for (int chunk = 0; chunk < BS / 4; ++chunk) {
        const int r0 = chunk * 4 + half * 2;
        v2f bm;
        bm.x = x[(size_t)(r0    ) * NC + colbase + ncol];
        bm.y = x[(size_t)(r0 + 1) * NC + colbase + ncol];
        // D = A*B + C  ->  every D row = column means of accumulated chunks
        c = __builtin_amdgcn_wmma_f32_16x16x4_f32(false, a, false, bm,
                                                  (short)0, c, false, false);
    }
    if (lane < 16) s_mean[colbase + lane] = c[0];   // D: VGPR0, lanes 0-15 = M=0,N=lane
    __syncthreads();

    for (int i = tid; i < BS * NC; i += blockDim.x) {
        const int j = i & (NC - 1);        // feature column
        float v = x[i];
        const float m = um[i], ac = ua[i];
        if (m < MASK_P) {
            if (ac < REPLACE_P)                 v = s_mean[j];
            else if (ac < REPLACE_P + RANDOM_P) v = x[(size_t)ridx[i] * NC + j];
        }
        out[i] = v;
    }
}

// =====================================================================
extern "C" void kernel_launch(void* const* d_in, const int* in_sizes, int n_in,
                              void* d_out, int out_size, void* d_ws, size_t ws_size,
                              hipStream_t stream)
{
    const float* x_ts       = (const float*)d_in[0];
    const float* x_ts_cat   = (const float*)d_in[1];
    const int*   x_cat      = (const int*)  d_in[2];
    const float* x_cont     = (const float*)d_in[3];
    const float* u_ts_mask  = (const float*)d_in[4];
    const float* u_ts_act   = (const float*)d_in[5];
    const int*   r_ts_idx   = (const int*)  d_in[6];
    const float* u_cts_mask = (const float*)d_in[7];
    const float* u_cts_act  = (const float*)d_in[8];
    const int*   r_cts_idx  = (const int*)  d_in[9];
    const float* u_cat_mask = (const float*)d_in[10];
    const float* u_cat_act  = (const float*)d_in[11];
    const int*   r_cat_val  = (const int*)  d_in[12];
    const float* u_cont_mask= (const float*)d_in[13];
    const float* u_cont_act = (const float*)d_in[14];
    const int*   r_cont_idx = (const int*)  d_in[15];

    const int BS = 64, C1 = 128, C2 = 256, T = 2048, NCAT = 32, NCONT = 64;

    float* out_ts   = (float*)d_out;
    float* out_cts  = out_ts  + (size_t)BS * C1 * T;
    float* out_cat  = out_cts + (size_t)BS * C2 * T;
    float* out_cont = out_cat + (size_t)BS * NCAT;

    dim3 g1(T / 1024, C1 / 32, BS);   // 2 x 4 x 64 = 512 blocks
    mask_ts_kernel<<<g1, 256, 0, stream>>>(x_ts, u_ts_mask, u_ts_act, r_ts_idx,
                                           out_ts, C1, T);

    dim3 g2(T / 1024, C2 / 32, BS);   // 2 x 8 x 64 = 1024 blocks
    mask_ts_kernel<<<g2, 256, 0, stream>>>(x_ts_cat, u_cts_mask, u_cts_act, r_cts_idx,
                                           out_cts, C2, T);

    mask_cat_kernel<<<(BS * NCAT + 255) / 256, 256, 0, stream>>>(
        x_cat, u_cat_mask, u_cat_act, r_cat_val, out_cat, BS * NCAT);

    mask_cont_kernel<<<1, 128, 0, stream>>>(
        x_cont, u_cont_mask, u_cont_act, r_cont_idx, out_cont);
}